// OriMHSLayer_39779987096204
// MI455X (gfx1250) — compile-verified
//
#include <hip/hip_runtime.h>

typedef float v2f __attribute__((ext_vector_type(2)));
typedef float v4f __attribute__((ext_vector_type(4)));
typedef float v8f __attribute__((ext_vector_type(8)));

#define DD 768   // input feature dim
#define HH 128   // hidden dim
#define LLBL 51  // label dim
#define LP 64    // padded label dim
#define SSEQ 1024
#define BB 2

// ---------------------------------------------------------------------------
// Kernel 1: fused projection.
//   One block (8 wave32) per 16-row m-tile of the flattened (B*S, D) input.
//   Stage 1: left/right = x @ W^T + bias  via v_wmma_f32_16x16x4_f32,
//            accumulator tiles written to LDS (re-stripe C/D layout -> A layout).
//   Stage 2: Lproj = (left) @ Wv^T + bv ;  RprojT[b][l][j] = (right) @ Wv^T
// ---------------------------------------------------------------------------
__global__ __launch_bounds__(256)
void proj_kernel(const float* __restrict__ x,
                 const float* __restrict__ Wu, const float* __restrict__ bu,
                 const float* __restrict__ Ww, const float* __restrict__ bw,
                 const float* __restrict__ Wv, const float* __restrict__ bv,
                 const float* __restrict__ bias,
                 float* __restrict__ Lproj, float* __restrict__ RprojT)
{
    __shared__ float tile[2][16 * HH];          // [which][m_local][h] = 16 KB

    const int tid  = threadIdx.x;
    const int wid  = tid >> 5;                  // wave id 0..7
    const int lane = tid & 31;
    const int n    = lane & 15;                 // column / row-within-halfwave
    const int hi   = lane >> 4;                 // half-wave select
    const int m0   = blockIdx.x * 16;           // this block's 16 rows of (B*S)

    // ---------------- stage 1: x(16xD) @ W^T(DxH) -> LDS(16xH), twice -------
    // 16 wave-tasks: which in {left,right} x 8 H-tiles; 2 tasks per wave.
    for (int t = wid * 2; t < wid * 2 + 2; ++t) {
        const int   which = t >> 3;             // 0 = Wu/left, 1 = Ww/right
        const int   ht    = t & 7;
        const int   h0    = ht * 16;
        const float* W    = which ? Ww : Wu;
        // fold bu+bias into left, bw into right (accumulator init; col = h)
        const float bval  = which ? bw[h0 + n] : (bu[h0 + n] + bias[h0 + n]);
        v8f acc;
        #pragma unroll
        for (int r = 0; r < 8; ++r) acc[r] = bval;

        const float* xrow = x + (size_t)(m0 + n) * DD + 2 * hi;  // A: row M=n
        const float* wrow = W + (size_t)(h0 + n) * DD + 2 * hi;  // B: col N=n
        #pragma unroll 4
        for (int d0 = 0; d0 < DD; d0 += 4) {
            v2f a = *(const v2f*)(xrow + d0);    // A[n][d0+2hi .. +1]
            v2f b = *(const v2f*)(wrow + d0);    // B[d0+2hi..][n]
            acc = __builtin_amdgcn_wmma_f32_16x16x4_f32(
                      false, a, false, b, (short)0, acc, false, false);
        }
        // C/D layout: acc[r] = element (M = r + 8*hi, N = n)
        #pragma unroll
        for (int r = 0; r < 8; ++r)
            tile[which][(r + 8 * hi) * HH + h0 + n] = acc[r];
    }
    __syncthreads();

    // ---------------- stage 2: LDS(16xH) @ Wv^T(HxLp) -----------------------
    // 8 wave-tasks: which x 4 L-tiles; exactly 1 per wave.
    {
        const int which = wid >> 2;
        const int l0    = (wid & 3) * 16;
        const int lcol  = l0 + n;               // B column = label index

        v8f acc;
        float cinit = (which == 0 && lcol < LLBL) ? bv[lcol] : 0.0f;
        #pragma unroll
        for (int r = 0; r < 8; ++r) acc[r] = cinit;

        const float* arow = &tile[which][n * HH + 2 * hi];
        #pragma unroll 4
        for (int k0 = 0; k0 < HH; k0 += 4) {
            v2f a = *(const v2f*)(arow + k0);   // A[n][k0+2hi .. +1] from LDS
            v2f b = {0.0f, 0.0f};               // pad l >= 51 with zeros
            if (lcol < LLBL)
                b = *(const v2f*)(Wv + (size_t)lcol * HH + k0 + 2 * hi);
            acc = __builtin_amdgcn_wmma_f32_16x16x4_f32(
                      false, a, false, b, (short)0, acc, false, false);
        }
        #pragma unroll
        for (int r = 0; r < 8; ++r) {
            const int gm = m0 + r + 8 * hi;     // global row in (B*S)
            if (which == 0) {
                Lproj[gm * LP + lcol] = acc[r];                 // [m][l] padded
            } else {
                const int bb = gm >> 10;
                const int j  = gm & (SSEQ - 1);
                RprojT[((bb * LP + lcol) << 10) + j] = acc[r];  // [b][l][j]
            }
        }
    }
}

// ---------------------------------------------------------------------------
// Kernel 2: the bandwidth-bound broadcast add (428 MB of write-once output).
//   One block per (b,i,l) output row of 1024 j's; float4 NT stores.
// ---------------------------------------------------------------------------
__global__ __launch_bounds__(256)
void bcast_kernel(const float* __restrict__ Lproj,
                  const float* __restrict__ RprojT,
                  float* __restrict__ out)
{
    const int bid = blockIdx.x;                 // (b*S + i)*51 + l
    const int m   = bid / LLBL;                 // b*S + i  (uniform -> SALU)
    const int l   = bid - m * LLBL;
    const int b   = m >> 10;

    const float Lval = Lproj[m * LP + l];       // uniform scalar per block
    const int   j    = threadIdx.x << 2;

    v4f r = *(const v4f*)(RprojT + ((b * LP + l) << 10) + j);  // L2-resident
    v4f o = r + Lval;

    __builtin_nontemporal_store(o, (v4f*)(out + ((size_t)bid << 10) + j));
}

extern "C" void kernel_launch(void* const* d_in, const int* in_sizes, int n_in,
                              void* d_out, int out_size, void* d_ws, size_t ws_size,
                              hipStream_t stream)
{
    const float* x    = (const float*)d_in[0];
    const float* Wu   = (const float*)d_in[1];
    const float* bu   = (const float*)d_in[2];
    const float* Ww   = (const float*)d_in[3];
    const float* bw   = (const float*)d_in[4];
    const float* Wv   = (const float*)d_in[5];
    const float* bv   = (const float*)d_in[6];
    const float* bias = (const float*)d_in[7];
    float* out = (float*)d_out;

    float* Lproj  = (float*)d_ws;               // (B*S) x 64   = 512 KB
    float* RprojT = Lproj + (BB * SSEQ) * LP;   // B x 64 x S   = 512 KB

    // 128 m-tiles of 16 rows; 256 threads (8 wave32) each.
    proj_kernel<<<BB * SSEQ / 16, 256, 0, stream>>>(
        x, Wu, bu, Ww, bw, Wv, bv, bias, Lproj, RprojT);

    // One block per (b, i, l) row; 256 threads x float4 = 1024 j's.
    bcast_kernel<<<BB * SSEQ * LLBL, 256, 0, stream>>>(Lproj, RprojT, out);
}